// PopulationLayer_42992622633706
// MI455X (gfx1250) — compile-verified
//
#include <hip/hip_runtime.h>

typedef float v2f __attribute__((ext_vector_type(2)));
typedef float v8f __attribute__((ext_vector_type(8)));

#define P_DIM 4096
#define B_DIM 256
#define GSPLIT 64                       // split-K row groups for the matvec
#define ROWS_PER_BLK (P_DIM / GSPLIT)   // 64 rows per partial block

// ---------------------------------------------------------------------------
// Phase 1: bandwidth-optimal split-K partial matvec over W (64 MB stream).
// partials[g][j] = sum_{i in row-block g} pop[i] * W[i][j]
// Each thread owns 4 consecutive columns (float4): a wave reads 512B
// contiguous per row -> fully coalesced streaming of W.
// ---------------------------------------------------------------------------
__global__ __launch_bounds__(256) void matvec_partial_kernel(
    const float* __restrict__ W, const float* __restrict__ pop,
    float* __restrict__ partials)
{
    const int col     = blockIdx.x * 1024 + threadIdx.x * 4;
    const int rowbase = blockIdx.y * ROWS_PER_BLK;

    float4 acc = make_float4(0.f, 0.f, 0.f, 0.f);

    for (int i = rowbase; i < rowbase + ROWS_PER_BLK; i += 4) {
        // pop chunk is uniform across the wave -> scalar-cached load
        const float4 p4 = *(const float4*)&pop[i];

        const float4 w0 = *(const float4*)&W[(size_t)(i + 0) * P_DIM + col];
        const float4 w1 = *(const float4*)&W[(size_t)(i + 1) * P_DIM + col];
        const float4 w2 = *(const float4*)&W[(size_t)(i + 2) * P_DIM + col];
        const float4 w3 = *(const float4*)&W[(size_t)(i + 3) * P_DIM + col];

        // gfx1250 prefetch path (global_prefetch_b8); keep default RT hint so
        // W stays resident in the 192MB L2 across graph replays.
        if (i + 8 < P_DIM)
            __builtin_prefetch(&W[(size_t)(i + 8) * P_DIM + col], 0, 3);

        acc.x = fmaf(p4.x, w0.x, acc.x); acc.y = fmaf(p4.x, w0.y, acc.y);
        acc.z = fmaf(p4.x, w0.z, acc.z); acc.w = fmaf(p4.x, w0.w, acc.w);
        acc.x = fmaf(p4.y, w1.x, acc.x); acc.y = fmaf(p4.y, w1.y, acc.y);
        acc.z = fmaf(p4.y, w1.z, acc.z); acc.w = fmaf(p4.y, w1.w, acc.w);
        acc.x = fmaf(p4.z, w2.x, acc.x); acc.y = fmaf(p4.z, w2.y, acc.y);
        acc.z = fmaf(p4.z, w2.z, acc.z); acc.w = fmaf(p4.z, w2.w, acc.w);
        acc.x = fmaf(p4.w, w3.x, acc.x); acc.y = fmaf(p4.w, w3.y, acc.y);
        acc.z = fmaf(p4.w, w3.z, acc.z); acc.w = fmaf(p4.w, w3.w, acc.w);
    }

    *(float4*)&partials[(size_t)blockIdx.y * P_DIM + col] = acc;
}

// ---------------------------------------------------------------------------
// Phase 2: reduce GSPLIT partials per column with V_WMMA_F32_16X16X4_F32.
// A = all-ones 16x4 tile (makes the K-slot layout irrelevant: D[m][n] =
// sum of all 4 B rows for column n, exactly). C accumulator is chained
// across g-blocks, so the whole reduction runs on the matrix pipe in fp32.
// Also applies the diagonal (self-connection) correction.
// ---------------------------------------------------------------------------
__global__ __launch_bounds__(128) void reduce_wmma_kernel(
    const float* __restrict__ partials, const float* __restrict__ W,
    const float* __restrict__ pop, float* __restrict__ lateral)
{
    const int lane = threadIdx.x & 31;
    const int wave = threadIdx.x >> 5;
    const int j0   = (blockIdx.x * 4 + wave) * 16;   // 16 columns per wave
    const int n    = lane & 15;
    const int j    = j0 + n;
    const int klo  = (lane < 16) ? 0 : 2;            // K-slot guess; harmless
                                                     // with all-ones A

    v2f a; a[0] = 1.0f; a[1] = 1.0f;                 // replicated ones rows
    v8f c = {};                                      // fp32 accumulator

    for (int g = 0; g < GSPLIT; g += 4) {
        v2f b;
        b[0] = partials[(size_t)(g + klo + 0) * P_DIM + j];
        b[1] = partials[(size_t)(g + klo + 1) * P_DIM + j];
        // 8 args: (neg_a, A, neg_b, B, c_mod, C, reuse_a, reuse_b)
        c = __builtin_amdgcn_wmma_f32_16x16x4_f32(
                false, a, false, b, (short)0, c, false, false);
    }

    // All D rows are identical (A rows identical); lanes 0-15 hold row M=0.
    if (lane < 16) {
        const float diag = pop[j] * W[(size_t)j * P_DIM + j];
        lateral[j] = c[0] - diag;
    }
}

// ---------------------------------------------------------------------------
// Phase 3: fused adaptive-threshold LIF step (elementwise over B x P).
// spikes = (hard - soft) + soft, matching the reference's evaluation order.
// ---------------------------------------------------------------------------
__device__ __forceinline__ float lif_one(float e, float vv, float th, float lat) {
    const float decay = 0.90483741803595952f;        // exp(-dt/tau), dt=1, tau=10
    const float vn   = fmaf(decay, vv, e - lat);
    const float u    = vn - th;                      // SURROGATE_BETA = 1
    const float soft = 1.0f / (1.0f + expf(-u));
    const float hard = (vn > th) ? 1.0f : 0.0f;
    return (hard - soft) + soft;
}

__global__ __launch_bounds__(256) void lif_step_kernel(
    const float* __restrict__ ext, const float* __restrict__ v,
    const float* __restrict__ thr, const float* __restrict__ lateral,
    float* __restrict__ out)
{
    const int t    = blockIdx.x * 256 + threadIdx.x;
    const int flat = t * 4;                          // float4 per thread
    const int j    = flat & (P_DIM - 1);             // column index (P = 4096)

    const float4 e  = *(const float4*)&ext[flat];
    const float4 vv = *(const float4*)&v[flat];
    const float4 th = *(const float4*)&thr[flat];
    const float4 lt = *(const float4*)&lateral[j];

    float4 r;
    r.x = lif_one(e.x, vv.x, th.x, lt.x);
    r.y = lif_one(e.y, vv.y, th.y, lt.y);
    r.z = lif_one(e.z, vv.z, th.z, lt.z);
    r.w = lif_one(e.w, vv.w, th.w, lt.w);

    *(float4*)&out[flat] = r;
}

// ---------------------------------------------------------------------------
extern "C" void kernel_launch(void* const* d_in, const int* in_sizes, int n_in,
                              void* d_out, int out_size, void* d_ws, size_t ws_size,
                              hipStream_t stream) {
    const float* ext = (const float*)d_in[0];   // [B, P] external_input
    const float* W   = (const float*)d_in[1];   // [P, P] lateral_weights
    const float* pop = (const float*)d_in[2];   // [P]    population_activity
    const float* v   = (const float*)d_in[3];   // [B, P] v
    const float* thr = (const float*)d_in[4];   // [B, P] threshold
    float* out = (float*)d_out;                 // [B, P] spikes

    float* partials = (float*)d_ws;                         // [GSPLIT, P]
    float* lateral  = partials + (size_t)GSPLIT * P_DIM;    // [P]

    // Phase 1: 4 column-blocks x 64 row-blocks, 256 threads each.
    dim3 g1(P_DIM / 1024, GSPLIT);
    matvec_partial_kernel<<<g1, 256, 0, stream>>>(W, pop, partials);

    // Phase 2: 64 blocks x 4 waves; each wave reduces 16 columns via WMMA.
    reduce_wmma_kernel<<<P_DIM / 64, 128, 0, stream>>>(partials, W, pop, lateral);

    // Phase 3: (256*4096)/4 float4 lanes / 256 threads = 1024 blocks.
    lif_step_kernel<<<(B_DIM * P_DIM) / (4 * 256), 256, 0, stream>>>(
        ext, v, thr, lateral, out);
}